// PointCloudAE_27247272525963
// MI455X (gfx1250) — compile-verified
//
#include <hip/hip_runtime.h>
#include <cstdint>

typedef __attribute__((ext_vector_type(2))) float v2f;
typedef __attribute__((ext_vector_type(8))) float v8f;

#define KNN 3

// ---------------------------------------------------------------------------
// prep: stage features as interleaved K-pairs, column-padded to Npad.
//   xt2[b][p][n] = (x[n][2p], x[n][2p+1])   (zero past C, zero rows past N)
//   sq[b][n]     = |x[n]|^2                 (1e30 for n >= N so padded columns
//                                            can never enter the top-3)
// ---------------------------------------------------------------------------
__global__ void prep_kernel(const float* __restrict__ x, v2f* __restrict__ xt2,
                            float* __restrict__ sq, int N, int Npad, int C, int KPAD) {
  const int n = blockIdx.x * blockDim.x + threadIdx.x;
  const int b = blockIdx.y;
  if (n >= Npad) return;
  const int P = KPAD >> 1;
  if (n < N) {
    const float* xr = x + ((size_t)b * N + n) * C;
    float s = 0.f;
    for (int p = 0; p < P; ++p) {
      const int k = 2 * p;
      v2f pr;
      pr.x = (k < C) ? xr[k] : 0.f;
      pr.y = (k + 1 < C) ? xr[k + 1] : 0.f;
      xt2[((size_t)b * P + p) * Npad + n] = pr;
      s += pr.x * pr.x + pr.y * pr.y;
    }
    sq[(size_t)b * Npad + n] = s;
  } else {
    v2f zz; zz.x = 0.f; zz.y = 0.f;
    for (int p = 0; p < P; ++p) xt2[((size_t)b * P + p) * Npad + n] = zz;
    sq[(size_t)b * Npad + n] = 1e30f;
  }
}

// ---------------------------------------------------------------------------
// knn: one wave owns 32 rows.  16x16 gram tiles via V_WMMA_F32_16X16X4_F32
// (two accumulators -> 32x16 keys per column tile).  Branch-free hot loop:
// unconditional b64 fragment loads, double-buffered LDS staging (one barrier
// per tile), branchless sorted top-3 insert (pure cndmask cascade).
// key = |x_j|^2 - 2<x_i,x_j> preserves the reference distance ordering.
// ---------------------------------------------------------------------------
template <int KPAD>
__global__ void knn_kernel(const v2f* __restrict__ xt2, const float* __restrict__ sq,
                           int* __restrict__ out_idx, int N, int Npad, int includeSelf) {
  constexpr int KS = KPAD / 4;   // WMMA k-steps
  constexpr int P = KPAD / 2;    // float2 pairs
  const int b = blockIdx.y;
  const int row0 = blockIdx.x * 32;
  const int lane = threadIdx.x;
  const int half = lane >> 4;    // K-half select (16x4 f32 A/B frag layout)
  const int l15 = lane & 15;
  const v2f* xtb = xt2 + (size_t)b * P * Npad;
  const float* sqb = sq + (size_t)b * Npad;

  __shared__ float dls[2][32][17];  // double buffer; +1 pad: conflict-free

  // Preload A fragments for rows row0..row0+31 (two 16-row halves).
  v2f a0[KS], a1[KS];
#pragma unroll
  for (int s = 0; s < KS; ++s) {
    const int p = s * 2 + half;
    a0[s] = xtb[(size_t)p * Npad + row0 + l15];
    a1[s] = xtb[(size_t)p * Npad + row0 + 16 + l15];
  }

  const float selfPen = includeSelf ? 0.f : 1e10f;
  const int gr = row0 + lane;    // row this lane owns in the scan phase

  float k0 = 1e30f, k1v = 1e30f, k2 = 1e30f;
  int i0 = 0, i1 = 0, i2 = 0;
  int pb = 0;

  for (int c0 = 0; c0 < Npad; c0 += 16) {
    const int col = c0 + l15;
    v8f acc0 = {};
    v8f acc1 = {};
#pragma unroll
    for (int s = 0; s < KS; ++s) {
      const v2f bf = xtb[(size_t)(s * 2 + half) * Npad + col];
      acc0 = __builtin_amdgcn_wmma_f32_16x16x4_f32(false, a0[s], false, bf,
                                                   (short)0, acc0, false, false);
      acc1 = __builtin_amdgcn_wmma_f32_16x16x4_f32(false, a1[s], false, bf,
                                                   (short)0, acc1, false, false);
    }
    const float sj = sqb[col];
#pragma unroll
    for (int v = 0; v < 8; ++v) {
      const int r0 = v + half * 8;               // C/D layout: M = v + 8*half
      dls[pb][r0][l15]      = fmaf(-2.f, acc0[v], sj);
      dls[pb][r0 + 16][l15] = fmaf(-2.f, acc1[v], sj);
    }
    __syncthreads();   // single barrier per tile (double-buffered staging)
#pragma unroll
    for (int c = 0; c < 16; ++c) {
      const int gc = c0 + c;
      const float key = dls[pb][lane][c] + ((gc == gr) ? selfPen : 0.f);
      const bool lt0 = key < k0;
      const bool lt1 = key < k1v;
      const bool lt2 = key < k2;
      // branchless sorted insert: update rank2 -> rank1 -> rank0
      k2  = lt1 ? k1v : (lt2 ? key : k2);
      i2  = lt1 ? i1  : (lt2 ? gc  : i2);
      k1v = lt0 ? k0  : (lt1 ? key : k1v);
      i1  = lt0 ? i0  : (lt1 ? gc  : i1);
      k0  = lt0 ? key : k0;
      i0  = lt0 ? gc  : i0;
    }
    pb ^= 1;
  }
  if (gr < N) {
    int* o = out_idx + ((size_t)b * N + gr) * KNN;
    o[0] = i0; o[1] = i1; o[2] = i2;
  }
}

// ---------------------------------------------------------------------------
// edge_conv: h[n] = max_k relu([x_i, x_j-x_i] @ W + b).  Weights in LDS.
// ---------------------------------------------------------------------------
__global__ void edge_conv_kernel(const float* __restrict__ x, const int* __restrict__ nidx,
                                 const float* __restrict__ W, const float* __restrict__ bias,
                                 float* __restrict__ out, int N, int C, int Cout) {
  __shared__ float sW[24 * 12];
  __shared__ float sB[12];
  const int b = blockIdx.y;
  const int tid = threadIdx.x;
  for (int t = tid; t < 2 * C * Cout; t += blockDim.x) sW[t] = W[t];
  if (tid < Cout) sB[tid] = bias[tid];
  __syncthreads();
  const int n = blockIdx.x * blockDim.x + tid;
  if (n >= N) return;

  float xi[12];
  const float* xr = x + ((size_t)b * N + n) * C;
  for (int c = 0; c < C; ++c) xi[c] = xr[c];
  float hmax[12];
  for (int co = 0; co < Cout; ++co) hmax[co] = -1e30f;

  const int* nb = nidx + ((size_t)b * N + n) * KNN;
  for (int kk = 0; kk < KNN; ++kk) {
    float xjl[12];
    const float* xj = x + ((size_t)b * N + nb[kk]) * C;
    for (int c = 0; c < C; ++c) xjl[c] = xj[c];
    for (int co = 0; co < Cout; ++co) {
      float acc = sB[co];
      for (int c = 0; c < C; ++c)
        acc += xi[c] * sW[c * Cout + co] + (xjl[c] - xi[c]) * sW[(C + c) * Cout + co];
      hmax[co] = fmaxf(hmax[co], fmaxf(acc, 0.f));
    }
  }
  float* orow = out + ((size_t)b * N + n) * Cout;
  for (int co = 0; co < Cout; ++co) orow[co] = hmax[co];
}

// ---------------------------------------------------------------------------
// fps: sequential farthest-point sampling.  One workgroup per batch element;
// mind[] lives in LDS; argmax with JAX first-index tie rule.
// ---------------------------------------------------------------------------
__global__ void fps_kernel(const float* __restrict__ x, int* __restrict__ out,
                           int N, int C, int M) {
  const int b = blockIdx.x;
  const int tid = threadIdx.x;
  const int T = blockDim.x;   // 256
  __shared__ float mind[2304];
  __shared__ float sval[256];
  __shared__ int sidx[256];
  __shared__ float xl[16];
  __shared__ int s_last;

  for (int n = tid; n < N; n += T) mind[n] = 1e10f;
  if (tid == 0) s_last = 0;
  __syncthreads();

  for (int t = 0; t < M; ++t) {
    const int last = s_last;
    if (tid == 0) out[(size_t)b * M + t] = last;
    if (tid < C) xl[tid] = x[((size_t)b * N + last) * C + tid];
    __syncthreads();

    float bv = -1.f; int bi = 0;
    for (int n = tid; n < N; n += T) {
      float d = 0.f;
      for (int c = 0; c < C; ++c) {
        const float df = x[((size_t)b * N + n) * C + c] - xl[c];
        d += df * df;
      }
      const float m = fminf(mind[n], d);
      mind[n] = m;
      bi = (m > bv) ? n : bi;          // strict > keeps lowest index on ties
      bv = fmaxf(bv, m);
    }
    sval[tid] = bv; sidx[tid] = bi;
    __syncthreads();
    for (int st = T >> 1; st > 0; st >>= 1) {
      if (tid < st) {
        const float v2 = sval[tid + st]; const int j2 = sidx[tid + st];
        const bool take = (v2 > sval[tid]) || (v2 == sval[tid] && j2 < sidx[tid]);
        sval[tid] = take ? v2 : sval[tid];
        sidx[tid] = take ? j2 : sidx[tid];
      }
      __syncthreads();
    }
    if (tid == 0) s_last = sidx[0];
    __syncthreads();
  }
}

__global__ void gather_kernel(const float* __restrict__ in, const int* __restrict__ idxs,
                              float* __restrict__ out, int N, int M, int C) {
  const int b = blockIdx.y;
  const int t = blockIdx.x * blockDim.x + threadIdx.x;
  if (t >= M) return;
  const int j = idxs[(size_t)b * M + t];
  const float* src = in + ((size_t)b * N + j) * C;
  float* dst = out + ((size_t)b * M + t) * C;
  for (int c = 0; c < C; ++c) dst[c] = src[c];
}

__global__ void linear_kernel(const float* __restrict__ in, const float* __restrict__ W,
                              const float* __restrict__ bias, float* __restrict__ out,
                              int Din, int Dout, int total) {
  const int t = blockIdx.x * blockDim.x + threadIdx.x;
  if (t >= total) return;
  const int b = t / Dout;
  const int o = t - b * Dout;
  float acc = bias[o];
  const float* ir = in + (size_t)b * Din;
  for (int i = 0; i < Din; ++i) acc += ir[i] * W[(size_t)i * Dout + o];
  out[t] = acc;
}

__global__ void unpool_kernel(const float* __restrict__ x, const int* __restrict__ nidx,
                              float* __restrict__ out, int m, int n, int C) {
  const int b = blockIdx.y;
  const int i = blockIdx.x * blockDim.x + threadIdx.x;
  if (i >= n) return;
  const int parent = (int)(((long long)i * m) / n);
  const int* nb = nidx + ((size_t)b * m + parent) * KNN;
  const float* x0 = x + ((size_t)b * m + nb[0]) * C;
  const float* x1 = x + ((size_t)b * m + nb[1]) * C;
  const float* x2 = x + ((size_t)b * m + nb[2]) * C;
  float* o = out + ((size_t)b * n + i) * C;
  const float inv3 = 1.f / 3.f;
  for (int c = 0; c < C; ++c) o[c] = (x0[c] + x1[c] + x2[c]) * inv3;
}

// ---------------------------------------------------------------------------
extern "C" void kernel_launch(void* const* d_in, const int* in_sizes, int n_in,
                              void* d_out, int out_size, void* d_ws, size_t ws_size,
                              hipStream_t stream) {
  (void)in_sizes; (void)n_in; (void)out_size; (void)ws_size;
  const float* x   = (const float*)d_in[0];
  const float* W1  = (const float*)d_in[1];  const float* b1  = (const float*)d_in[2];
  const float* W2  = (const float*)d_in[3];  const float* b2  = (const float*)d_in[4];
  const float* W3  = (const float*)d_in[5];  const float* b3  = (const float*)d_in[6];
  const float* Wl1 = (const float*)d_in[7];  const float* bl1 = (const float*)d_in[8];
  const float* Wl2 = (const float*)d_in[9];  const float* bl2 = (const float*)d_in[10];
  const float* W4  = (const float*)d_in[11]; const float* b4  = (const float*)d_in[12];
  const float* W5  = (const float*)d_in[13]; const float* b5  = (const float*)d_in[14];
  const float* W6  = (const float*)d_in[15]; const float* b6  = (const float*)d_in[16];

  const int B = 32, N1 = 2300, N2 = 500, N3 = 50;
  const int Npad1 = 2304;  // (N+31)&~31

  char* base = (char*)d_ws;
  size_t off = 0;
  auto alloc = [&](size_t bytes) -> void* {
    void* p = base + off;
    off = (off + bytes + 255) & ~(size_t)255;
    return p;
  };
  v2f*   xt   = (v2f*)  alloc(sizeof(v2f)   * (size_t)B * 6 * Npad1);  // up to 6 pairs
  float* sqb  = (float*)alloc(sizeof(float) * (size_t)B * Npad1);
  int*   kidx = (int*)  alloc(sizeof(int)   * (size_t)B * N1 * KNN);
  float* h1   = (float*)alloc(sizeof(float) * (size_t)B * N1 * 9);
  int*   f500 = (int*)  alloc(sizeof(int)   * (size_t)B * N2);
  float* h1s  = (float*)alloc(sizeof(float) * (size_t)B * N2 * 9);
  float* h2   = (float*)alloc(sizeof(float) * (size_t)B * N2 * 12);
  int*   f50  = (int*)  alloc(sizeof(int)   * (size_t)B * N3);
  float* h2s  = (float*)alloc(sizeof(float) * (size_t)B * N3 * 12);
  float* h3   = (float*)alloc(sizeof(float) * (size_t)B * N3 * 12);
  float* z    = (float*)alloc(sizeof(float) * (size_t)B * 30);
  float* hd   = (float*)alloc(sizeof(float) * (size_t)B * 600);
  float* h4   = (float*)alloc(sizeof(float) * (size_t)B * N3 * 12);
  float* u1   = (float*)alloc(sizeof(float) * (size_t)B * N2 * 12);
  float* h5   = (float*)alloc(sizeof(float) * (size_t)B * N2 * 9);
  float* u2   = (float*)alloc(sizeof(float) * (size_t)B * N1 * 9);

  auto knn = [&](const float* feat, int N, int C, int KPAD, int inclSelf) {
    const int Npad = (N + 31) & ~31;
    dim3 gp((Npad + 255) / 256, B);
    prep_kernel<<<gp, 256, 0, stream>>>(feat, xt, sqb, N, Npad, C, KPAD);
    dim3 gk(Npad / 32, B);
    if (KPAD == 4) knn_kernel<4><<<gk, 32, 0, stream>>>(xt, sqb, kidx, N, Npad, inclSelf);
    else           knn_kernel<12><<<gk, 32, 0, stream>>>(xt, sqb, kidx, N, Npad, inclSelf);
  };
  auto econv = [&](const float* feat, const float* Wp, const float* bp, float* outp,
                   int N, int C, int Cout) {
    dim3 g((N + 255) / 256, B);
    edge_conv_kernel<<<g, 256, 0, stream>>>(feat, kidx, Wp, bp, outp, N, C, Cout);
  };

  // ---------------- encoder ----------------
  knn(x, N1, 3, 4, 0);
  econv(x, W1, b1, h1, N1, 3, 9);                                   // [B,2300,9]
  fps_kernel<<<B, 256, 0, stream>>>(h1, f500, N1, 9, N2);
  gather_kernel<<<dim3((N2 + 255) / 256, B), 256, 0, stream>>>(h1, f500, h1s, N1, N2, 9);
  knn(h1s, N2, 9, 12, 0);
  econv(h1s, W2, b2, h2, N2, 9, 12);                                // [B,500,12]
  fps_kernel<<<B, 256, 0, stream>>>(h2, f50, N2, 12, N3);
  gather_kernel<<<dim3((N3 + 255) / 256, B), 256, 0, stream>>>(h2, f50, h2s, N2, N3, 12);
  knn(h2s, N3, 12, 12, 0);
  econv(h2s, W3, b3, h3, N3, 12, 12);                               // [B,50,12]
  linear_kernel<<<(B * 30 + 255) / 256, 256, 0, stream>>>(h3, Wl1, bl1, z, 600, 30, B * 30);

  // ---------------- decoder ----------------
  linear_kernel<<<(B * 600 + 255) / 256, 256, 0, stream>>>(z, Wl2, bl2, hd, 30, 600, B * 600);
  knn(hd, N3, 12, 12, 0);
  econv(hd, W4, b4, h4, N3, 12, 12);                                // [B,50,12]
  knn(h4, N3, 12, 12, 1);
  unpool_kernel<<<dim3((N2 + 255) / 256, B), 256, 0, stream>>>(h4, kidx, u1, N3, N2, 12);
  knn(u1, N2, 12, 12, 0);
  econv(u1, W5, b5, h5, N2, 12, 9);                                 // [B,500,9]
  knn(h5, N2, 9, 12, 1);
  unpool_kernel<<<dim3((N1 + 255) / 256, B), 256, 0, stream>>>(h5, kidx, u2, N2, N1, 9);
  knn(u2, N1, 9, 12, 0);
  econv(u2, W6, b6, (float*)d_out, N1, 9, 3);                       // [B,2300,3]
}